// SAGEPoolAggr_81209241632839
// MI455X (gfx1250) — compile-verified
//
#include <hip/hip_runtime.h>
#include <hip/hip_bf16.h>

typedef __attribute__((ext_vector_type(2))) float v2f;
typedef __attribute__((ext_vector_type(8))) float v8f;

#define D_FEAT 128

// ---------------------------------------------------------------------------
// Kernel 1: feat = relu(x @ W + bias), f32 precision via V_WMMA_F32_16X16X4_F32
// Block = 256 threads = 8 waves. Block b -> rows [16b, 16b+16). Wave w -> cols
// [16w, 16w+16). K-loop: 32 steps of K=4.
// ---------------------------------------------------------------------------
__global__ void __launch_bounds__(256)
sage_gemm_bias_relu(const float* __restrict__ x,
                    const float* __restrict__ w,
                    const float* __restrict__ bias,
                    float* __restrict__ feat) {
    const int lane  = threadIdx.x & 31;
    const int wave  = threadIdx.x >> 5;      // 0..7
    const int tileM = blockIdx.x << 4;       // row base of this block
    const int tileN = wave << 4;             // col base of this wave
    const int half  = lane >> 4;             // 0: lanes 0-15, 1: lanes 16-31
    const int l16   = lane & 15;

    // A-matrix 16x4 f32 layout: lane l16 holds row M=l16; lanes 0-15 carry
    // K={0,1}, lanes 16-31 carry K={2,3}.  B-matrix 4x16 is the mirror:
    // lane l16 holds col N=l16 with the same K split.
    const int   kOff = half << 1;
    const float* xrow = x + (size_t)(tileM + l16) * D_FEAT;
    const float* wcol = w + tileN + l16;

    v8f c = {};
    #pragma unroll 4
    for (int k = 0; k < D_FEAT; k += 4) {
        v2f a, b;
        a.x = xrow[k + kOff + 0];
        a.y = xrow[k + kOff + 1];
        b.x = wcol[(size_t)(k + kOff + 0) * D_FEAT];
        b.y = wcol[(size_t)(k + kOff + 1) * D_FEAT];
        // D = A*B + C  (8 args: neg_a, A, neg_b, B, c_mod, C, reuse_a, reuse_b)
        c = __builtin_amdgcn_wmma_f32_16x16x4_f32(
                false, a, false, b, (short)0, c, false, false);
    }

    // C/D layout: lanes 0-15 -> N=l16, M=vgpr r; lanes 16-31 -> M=vgpr r + 8.
    const int   n  = tileN + l16;
    const float bn = bias[n];
    #pragma unroll
    for (int r = 0; r < 8; ++r) {
        const int   m = tileM + r + (half << 3);
        float v = c[r] + bn;
        v = v > 0.0f ? v : 0.0f;               // ReLU
        feat[(size_t)m * D_FEAT + n] = v;
    }
}

// ---------------------------------------------------------------------------
// Kernel 2: zero the aggregation output (0x00000000 == +0.0f; also the fill
// value for empty segments, matching reference's maximum(agg, 0)).
// ---------------------------------------------------------------------------
__global__ void __launch_bounds__(256)
sage_zero(float4* __restrict__ out, int n4) {
    const int i = blockIdx.x * blockDim.x + threadIdx.x;
    if (i < n4) out[i] = make_float4(0.f, 0.f, 0.f, 0.f);
}

// ---------------------------------------------------------------------------
// Kernel 3: per-edge gather + segment max.  One wave32 per edge; each lane
// moves 4 contiguous floats (float4 / b128 load) and does 4 u32 atomic maxes.
// All feat values are >= 0 post-ReLU, so unsigned-int compare == float
// compare, and the zero-initialized output is the correct identity.
// feat (20.5 MB) and out (20.5 MB) both live in the 192 MB L2, so the random
// gather/scatter stays on-chip.
// ---------------------------------------------------------------------------
__global__ void __launch_bounds__(256)
sage_edge_max(const int* __restrict__ row,
              const int* __restrict__ col,
              const float* __restrict__ feat,
              unsigned int* __restrict__ out,
              int n_edges) {
    const int e    = (int)((blockIdx.x * (unsigned)blockDim.x + threadIdx.x) >> 5);
    const int lane = threadIdx.x & 31;
    if (e >= n_edges) return;

    const int r = row[e];
    const int c = col[e];

    const float4 v = *((const float4*)(feat + (size_t)c * D_FEAT) + lane);
    unsigned int* dst = out + (size_t)r * D_FEAT + (lane << 2);

    atomicMax(dst + 0, __float_as_uint(v.x));
    atomicMax(dst + 1, __float_as_uint(v.y));
    atomicMax(dst + 2, __float_as_uint(v.z));
    atomicMax(dst + 3, __float_as_uint(v.w));
}

// ---------------------------------------------------------------------------
// Launcher. Inputs (setup_inputs order): x [40000*128] f32, edge_index [2*E]
// int, weight [128*128] f32, bias [128] f32.  Output: [40000*128] f32.
// Workspace: feat buffer (40000*128*4 = 20.5 MB).
// ---------------------------------------------------------------------------
extern "C" void kernel_launch(void* const* d_in, const int* in_sizes, int n_in,
                              void* d_out, int out_size, void* d_ws, size_t ws_size,
                              hipStream_t stream) {
    const float* x     = (const float*)d_in[0];
    const int*   edges = (const int*)d_in[1];   // [2, E] flattened
    const float* w     = (const float*)d_in[2];
    const float* bias  = (const float*)d_in[3];

    float*        out   = (float*)d_out;
    float*        feat  = (float*)d_ws;

    const int n_nodes = in_sizes[0] / D_FEAT;   // 40000
    const int n_edges = in_sizes[1] / 2;        // 640000
    const int* row = edges;                      // edge_index[0]
    const int* col = edges + n_edges;            // edge_index[1]

    // 1) feat = relu(x @ W + b)   (2500 blocks of 256 = 8 waves, WMMA f32)
    sage_gemm_bias_relu<<<n_nodes / 16, 256, 0, stream>>>(x, w, bias, feat);

    // 2) out = 0
    const int n4 = out_size / 4;
    sage_zero<<<(n4 + 255) / 256, 256, 0, stream>>>((float4*)out, n4);

    // 3) segment max over edges (one wave per edge)
    const long long threads = (long long)n_edges * 32;
    sage_edge_max<<<(int)((threads + 255) / 256), 256, 0, stream>>>(
        row, col, feat, (unsigned int*)out, n_edges);
}